// LayerGT_60017872995081
// MI455X (gfx1250) — compile-verified
//
#include <hip/hip_runtime.h>
#include <math.h>

// Problem constants (match the reference)
#define N_NODES 50000
#define N_EDGES 800000
#define KGRP    8
#define CCH     64
#define DDIM    512          // KGRP * CCH
#define NODE_TILES (N_NODES / 16)   // 3125 exactly
#define COL_TILES  (DDIM / 16)      // 32

typedef float v2f __attribute__((ext_vector_type(2)));
typedef float v8f __attribute__((ext_vector_type(8)));

// ---------------------------------------------------------------------------
// Zero fill (float4 granularity). Harness poisons d_out/d_ws with 0xAA, so we
// must zero our accumulators every call (deterministic across graph replays).
// ---------------------------------------------------------------------------
__global__ void zero_f4_kernel(float4* __restrict__ p, int n4) {
    int i = blockIdx.x * blockDim.x + threadIdx.x;
    if (i < n4) p[i] = make_float4(0.f, 0.f, 0.f, 0.f);
}

// ---------------------------------------------------------------------------
// deg[dst] += 1 for every edge (float atomics; deg later gets +1 self-loop)
// ---------------------------------------------------------------------------
__global__ void degree_kernel(const long long* __restrict__ ei, float* __restrict__ deg) {
    int e = blockIdx.x * blockDim.x + threadIdx.x;
    if (e < N_EDGES) {
        int dst = (int)ei[N_EDGES + e];
        atomicAdd(&deg[dst], 1.0f);
    }
}

__global__ void dinv_kernel(const float* __restrict__ deg, float* __restrict__ dinv) {
    int n = blockIdx.x * blockDim.x + threadIdx.x;
    if (n < N_NODES) dinv[n] = rsqrtf(deg[n] + 1.0f);
}

// ---------------------------------------------------------------------------
// Per-group transform xt_k = x_k @ W_k^T via V_WMMA_F32_16X16X4_F32.
// One wave per 16-node x 16-outchannel tile; 16 chained WMMAs cover K=64.
// Input channel c' of group k lives at feature index c'*8 + k (reshape(N,C,K)).
// Output xt is stored in the *concatenated* layout: column k*64 + c.
// ---------------------------------------------------------------------------
__global__ void __launch_bounds__(256)
transform_wmma_kernel(const float* __restrict__ x,
                      const float* __restrict__ convW,   // [K][64][64]
                      float* __restrict__ xt) {          // [N][512]
    const int wave = blockIdx.x * (blockDim.x >> 5) + (threadIdx.x >> 5);
    const int lane = threadIdx.x & 31;
    const int tcol  = wave & (COL_TILES - 1);   // 0..31
    const int tnode = wave >> 5;                // 0..3124
    if (tnode >= NODE_TILES) return;            // never trips (exact division)

    const int k     = tcol >> 2;                // group 0..7
    const int cbase = (tcol & 3) << 4;          // 0,16,32,48
    const int half  = lane >> 4;                // K-half selector
    const int lrow  = lane & 15;                // A: M row / B: N col

    // A fragment source: x[node, (kk)*8 + k]
    const float* xrow = x + (size_t)(tnode * 16 + lrow) * DDIM + k;
    // B fragment source: W_k[c, kk]  (xt = x_k @ W_k^T)
    const float* wrow = convW + ((size_t)k * CCH + (cbase + lrow)) * CCH;

    v8f acc = {0.f, 0.f, 0.f, 0.f, 0.f, 0.f, 0.f, 0.f};
#pragma unroll
    for (int kc = 0; kc < 16; ++kc) {
        const int kk = kc * 4 + half * 2;       // this lane-half's K pair
        v2f A = { xrow[(size_t)(kk + 0) * KGRP], xrow[(size_t)(kk + 1) * KGRP] };
        v2f B = { wrow[kk + 0],                  wrow[kk + 1] };
        acc = __builtin_amdgcn_wmma_f32_16x16x4_f32(
            /*neg_a=*/false, A, /*neg_b=*/false, B,
            /*c_mod=*/(short)0, acc, /*reuse_a=*/false, /*reuse_b=*/false);
    }

    // D layout: VGPR v -> row M = v + 8*half, col N = lrow
    float* orow = xt + (size_t)(tnode * 16 + half * 8) * DDIM + (k * CCH + cbase + lrow);
#pragma unroll
    for (int v = 0; v < 8; ++v) orow[(size_t)v * DDIM] = acc[v];
}

// ---------------------------------------------------------------------------
// Edge aggregation: out[dst,:] += dinv[src]*dinv[dst] * xt[src,:]
// One wave per edge; b128 gathers + global_atomic_add_f32 scatters.
// ---------------------------------------------------------------------------
__global__ void __launch_bounds__(256)
scatter_kernel(const long long* __restrict__ ei,
               const float* __restrict__ dinv,
               const float* __restrict__ xt,
               float* __restrict__ out) {
    const int wid  = blockIdx.x * (blockDim.x >> 5) + (threadIdx.x >> 5);
    const int lane = threadIdx.x & 31;
    if (wid >= N_EDGES) return;

    const int src = (int)ei[wid];
    const int dst = (int)ei[N_EDGES + wid];
    const float nrm = dinv[src] * dinv[dst];

    const float4* srow = (const float4*)(xt + (size_t)src * DDIM);
    float* drow = out + (size_t)dst * DDIM;
#pragma unroll
    for (int it = 0; it < 4; ++it) {
        const int c4 = it * 32 + lane;          // float4 index 0..127
        float4 v = srow[c4];
        const int c = c4 * 4;
        atomicAdd(&drow[c + 0], v.x * nrm);
        atomicAdd(&drow[c + 1], v.y * nrm);
        atomicAdd(&drow[c + 2], v.z * nrm);
        atomicAdd(&drow[c + 3], v.w * nrm);
    }
}

// ---------------------------------------------------------------------------
// Finalize: out = leaky(leaky(out + dinv^2 * xt + bias)).
// Bias index k*64+c == feature index d, so conv_b reads are direct.
// Softmax over a length-1 axis == 1, so the FC/gate stage vanishes exactly.
// ---------------------------------------------------------------------------
__device__ __forceinline__ float leaky2(float v) {
    v = (v >= 0.f) ? v : 0.01f * v;   // inner leaky_relu
    v = (v >= 0.f) ? v : 0.01f * v;   // outer leaky_relu
    return v;
}

__global__ void __launch_bounds__(256)
finalize_kernel(const float* __restrict__ xt,
                const float* __restrict__ dinv,
                const float* __restrict__ bias,   // [512]
                float* __restrict__ out) {
    const int i4 = blockIdx.x * blockDim.x + threadIdx.x;   // float4 index
    if (i4 >= N_NODES * DDIM / 4) return;
    const int n  = i4 >> 7;           // 128 float4 per row
    const int c4 = i4 & 127;

    const float di = dinv[n];
    const float d2 = di * di;
    float4 av = ((const float4*)out)[i4];
    float4 tv = ((const float4*)xt)[i4];
    float4 bv = ((const float4*)bias)[c4];

    float4 r;
    r.x = leaky2(av.x + d2 * tv.x + bv.x);
    r.y = leaky2(av.y + d2 * tv.y + bv.y);
    r.z = leaky2(av.z + d2 * tv.z + bv.z);
    r.w = leaky2(av.w + d2 * tv.w + bv.w);
    ((float4*)out)[i4] = r;
}

// ---------------------------------------------------------------------------
extern "C" void kernel_launch(void* const* d_in, const int* in_sizes, int n_in,
                              void* d_out, int out_size, void* d_ws, size_t ws_size,
                              hipStream_t stream) {
    (void)in_sizes; (void)n_in; (void)out_size; (void)ws_size;

    const float*     x     = (const float*)d_in[0];
    const long long* ei    = (const long long*)d_in[1];   // int64 [2, E]
    const float*     convW = (const float*)d_in[2];       // [8,64,64]
    const float*     convb = (const float*)d_in[3];       // [8,64]
    // d_in[4] (W1), d_in[5] (W2) unused: softmax over a length-1 axis => gate==1.

    float* out = (float*)d_out;                           // [N, 512] f32

    // Workspace layout (floats), 16B-aligned sections:
    //   deg  : N_NODES     (padded to 50176)
    //   dinv : N_NODES     (padded to 50176)
    //   xt   : N_NODES*512
    float* ws   = (float*)d_ws;
    float* deg  = ws;
    float* dinv = ws + 50176;
    float* xt   = ws + 2 * 50176;

    const int B = 256;

    // 1) zero accumulators (deg + out-as-agg)
    zero_f4_kernel<<<(N_NODES / 4 + B - 1) / B, B, 0, stream>>>((float4*)deg, N_NODES / 4);
    zero_f4_kernel<<<(N_NODES * DDIM / 4 + B - 1) / B, B, 0, stream>>>((float4*)out,
                                                                       N_NODES * DDIM / 4);
    // 2) degrees
    degree_kernel<<<(N_EDGES + B - 1) / B, B, 0, stream>>>(ei, deg);
    // 3) dinv = (deg+1)^-1/2
    dinv_kernel<<<(N_NODES + B - 1) / B, B, 0, stream>>>(deg, dinv);
    // 4) per-group transform via f32 WMMA (8 waves/block, 1 tile/wave)
    transform_wmma_kernel<<<(NODE_TILES * COL_TILES) / 8, B, 0, stream>>>(x, convW, xt);
    // 5) edge gather/scatter (1 wave/edge)
    scatter_kernel<<<N_EDGES / 8, B, 0, stream>>>(ei, dinv, xt, out);
    // 6) self-loop + bias + double leaky_relu (gate == 1)
    finalize_kernel<<<(N_NODES * DDIM / 4 + B - 1) / B, B, 0, stream>>>(xt, dinv, convb, out);
}